// LangevinSampler_30605936951680
// MI455X (gfx1250) — compile-verified
//
#include <hip/hip_runtime.h>
#include <stdint.h>

#define Bdim 4
#define Sdim 512
#define Edim 64
#define Vdim 50257
#define KSEL 250
#define NBINS 8192
#define LISTCAP 1024
#define TOPK_THREADS 256

typedef __attribute__((ext_vector_type(2))) float v2f;
typedef __attribute__((ext_vector_type(8))) float v8f;

// LDS layout for topk kernel (dynamic shared)
#define ROW_BYTES   (Vdim * 4)                      // 201028
#define HIST_OFF    (ROW_BYTES)                     // 8192 bins
#define CVAL_OFF    (HIST_OFF + NBINS * 4)
#define CIDX_OFF    (CVAL_OFF + LISTCAP * 4)
#define PSUM_OFF    (CIDX_OFF + LISTCAP * 4)
#define VARS_OFF    (PSUM_OFF + TOPK_THREADS * 4)
#define SMEM_BYTES  (VARS_OFF + 16)                 // ~243 KB < 320 KB WGP LDS

__device__ __forceinline__ unsigned orderKey(float f) {
    unsigned u = __float_as_uint(f);
    // monotonic map: larger float -> larger unsigned
    return (u & 0x80000000u) ? ~u : (u | 0x80000000u);
}

// One workgroup per (b,s) row. Stage the 201KB row into LDS (single HBM pass),
// radix-select the rank-K threshold, compact candidates, bitonic-sort, emit
// sorted top-K ids (descending value, ascending index on ties).
__global__ void topk_kernel(const float* __restrict__ logits,
                            int* __restrict__ out_ids) {
    const int row = blockIdx.x;
    const int tid = threadIdx.x;
    extern __shared__ unsigned char smem[];
    unsigned* rowu = (unsigned*)smem;
    unsigned* hist = (unsigned*)(smem + HIST_OFF);
    unsigned* cval = (unsigned*)(smem + CVAL_OFF);
    int*      cidx = (int*)(smem + CIDX_OFF);
    unsigned* psum = (unsigned*)(smem + PSUM_OFF);
    unsigned* vars = (unsigned*)(smem + VARS_OFF);  // [0]=threshold bin, [1]=cand count

    for (int i = tid; i < NBINS; i += TOPK_THREADS) hist[i] = 0u;
    if (tid < 2) vars[tid] = 0u;
    __syncthreads();

    // Pass 1: stream row from global -> LDS, build 13-bit histogram.
    // Unrolled x4 so each lane keeps 4 global loads in flight (HBM-bound pass).
    const float* src = logits + (size_t)row * Vdim;
    int i = tid;
    for (; i + 3 * TOPK_THREADS < Vdim; i += 4 * TOPK_THREADS) {
        float f0 = src[i];
        float f1 = src[i + TOPK_THREADS];
        float f2 = src[i + 2 * TOPK_THREADS];
        float f3 = src[i + 3 * TOPK_THREADS];
        unsigned u0 = orderKey(f0), u1 = orderKey(f1);
        unsigned u2 = orderKey(f2), u3 = orderKey(f3);
        rowu[i] = u0;
        rowu[i + TOPK_THREADS] = u1;
        rowu[i + 2 * TOPK_THREADS] = u2;
        rowu[i + 3 * TOPK_THREADS] = u3;
        atomicAdd(&hist[u0 >> 19], 1u);
        atomicAdd(&hist[u1 >> 19], 1u);
        atomicAdd(&hist[u2 >> 19], 1u);
        atomicAdd(&hist[u3 >> 19], 1u);
    }
    for (; i < Vdim; i += TOPK_THREADS) {
        unsigned u = orderKey(src[i]);
        rowu[i] = u;
        atomicAdd(&hist[u >> 19], 1u);
    }
    __syncthreads();

    // Parallel suffix accumulation: each thread sums 32 bins
    {
        unsigned s = 0;
        const int base = tid * 32;
        #pragma unroll
        for (int b = 0; b < 32; ++b) s += hist[base + b];
        psum[tid] = s;
    }
    __syncthreads();

    // Serial scan (short) on thread 0: find threshold bin T such that
    // count(bins > T) < K <= count(bins >= T)
    if (tid == 0) {
        unsigned acc = 0;
        int seg = 0;
        for (int t = TOPK_THREADS - 1; t >= 0; --t) {
            if (acc + psum[t] >= (unsigned)KSEL) { seg = t; break; }
            acc += psum[t];
        }
        unsigned acc2 = acc;
        unsigned T = (unsigned)(seg * 32);
        for (int b = seg * 32 + 31; b >= seg * 32; --b) {
            acc2 += hist[b];
            if (acc2 >= (unsigned)KSEL) { T = (unsigned)b; break; }
        }
        vars[0] = T;
    }
    __syncthreads();
    const unsigned T = vars[0];

    // Pass 2 (from LDS): compact candidates with bucket >= T
    for (int j = tid; j < Vdim; j += TOPK_THREADS) {
        unsigned u = rowu[j];
        if ((u >> 19) >= T) {
            unsigned pos = atomicAdd(&vars[1], 1u);
            if (pos < LISTCAP) { cval[pos] = u; cidx[pos] = j; }
        }
    }
    __syncthreads();
    unsigned n = vars[1];
    if (n > LISTCAP) n = LISTCAP;
    for (int j = tid; j < LISTCAP; j += TOPK_THREADS) {
        if ((unsigned)j >= n) { cval[j] = 0u; cidx[j] = 0x7FFFFFFF; }
    }
    __syncthreads();

    // Bitonic sort, descending by key (value), ascending index on ties
    for (unsigned k = 2; k <= LISTCAP; k <<= 1) {
        for (unsigned j = k >> 1; j > 0; j >>= 1) {
            for (unsigned p = tid; p < LISTCAP; p += TOPK_THREADS) {
                unsigned l = p ^ j;
                if (l > p) {
                    unsigned ua = cval[p], ub = cval[l];
                    int ia = cidx[p], ib = cidx[l];
                    bool aWorse = (ua < ub) || (ua == ub && ia > ib);
                    bool desc = ((p & k) == 0);
                    if (desc ? aWorse : !aWorse) {
                        cval[p] = ub; cval[l] = ua;
                        cidx[p] = ib; cidx[l] = ia;
                    }
                }
            }
            __syncthreads();
        }
    }

    for (int t = tid; t < KSEL; t += TOPK_THREADS)
        out_ids[(size_t)row * KSEL + t] = cidx[t];
}

// One wave32 per (b,s) row. For each 16-candidate tile, gather W rows as the
// WMMA A matrix (f32 16x16x4), with gx in B column 0 and cur_bias in column 1.
// The B fragments are per-lane constants -> precomputed once, branch-free, so
// the inner loop is purely: global_load_b64 (A) + FMA (||w||^2) + v_wmma.
__global__ void proposal_kernel(const float* __restrict__ gx,
                                const float* __restrict__ bias,
                                const float* __restrict__ W,
                                const int* __restrict__ ids,
                                float* __restrict__ outvals) {
    const int row  = blockIdx.x;
    const int lane = threadIdx.x;   // 0..31
    __shared__ float sgx[Edim];
    __shared__ float sb[Edim];
    __shared__ float dmat[256];
    __shared__ float nwarr[16];

    const float* g  = gx   + (size_t)row * Edim;
    const float* bb = bias + (size_t)row * Edim;
    float g0 = g[lane],  g1 = g[lane + 32];
    float b0 = bb[lane], b1 = bb[lane + 32];
    sgx[lane] = g0; sgx[lane + 32] = g1;
    sb[lane]  = b0; sb[lane + 32]  = b1;

    // row scalars: gx.bias and ||bias||^2, wave-wide reduction
    float pd = g0 * b0 + g1 * b1;
    float pn = b0 * b0 + b1 * b1;
    #pragma unroll
    for (int off = 16; off > 0; off >>= 1) {
        pd += __shfl_xor(pd, off, 32);
        pn += __shfl_xor(pn, off, 32);
    }
    __syncthreads();

    const int   m  = lane & 15;                // candidate within tile / B column
    const int   kb = (lane < 16) ? 0 : 2;      // K sub-offset per lane half
    const float dotgb = pd, nb = pn;
    const float inv_step = 10.0f;              // 1/STEP_SIZE

    // Precompute all 16 B fragments (4x16: col0=gx, col1=bias, rest 0).
    // Branch-free select-by-multiply: keeps EXEC untouched, no LDS in hot loop.
    const float s0 = (m == 0) ? 1.0f : 0.0f;
    const float s1 = (m == 1) ? 1.0f : 0.0f;
    v2f bfrag[16];
    #pragma unroll
    for (int k4 = 0; k4 < 16; ++k4) {
        const int k = k4 * 4 + kb;
        v2f bv;
        bv.x = sgx[k]     * s0 + sb[k]     * s1;
        bv.y = sgx[k + 1] * s0 + sb[k + 1] * s1;
        bfrag[k4] = bv;
    }

    for (int tile = 0; tile < 16; ++tile) {
        const int kidx = tile * 16 + m;
        const int cid  = (kidx < KSEL) ? ids[(size_t)row * KSEL + kidx] : 0;
        const float* wrow = W + (size_t)cid * Edim;

        v8f c = {};
        float sq = 0.0f;
        #pragma unroll
        for (int k4 = 0; k4 < 16; ++k4) {
            // A (16x4 f32): lanes 0-15 hold K=k,k+1 of row m; lanes 16-31 K=k+2,k+3
            v2f a = *(const v2f*)(wrow + k4 * 4 + kb);
            sq += a.x * a.x + a.y * a.y;
            c = __builtin_amdgcn_wmma_f32_16x16x4_f32(
                    false, a, false, bfrag[k4], (short)0, c, false, false);
        }
        // ||w||^2: combine the two lane halves of the row
        float nw = sq + __shfl_xor(sq, 16, 32);
        if (lane < 16) nwarr[m] = nw;

        // Spill D (16x16) to LDS to remap layout: lane<16 rows 0-7, lane>=16 rows 8-15
        #pragma unroll
        for (int r = 0; r < 8; ++r) {
            int mr = (lane < 16) ? r : (8 + r);
            dmat[mr * 16 + m] = c[r];
        }
        __syncthreads();

        if (lane < 16 && kidx < KSEL) {
            float t11 = dmat[m * 16 + 0];   // gx . w
            float t22 = dmat[m * 16 + 1];   // bias . w
            float val = -(0.5f * (t11 - dotgb) +
                          (nwarr[m] - 2.0f * t22 + nb) * inv_step);
            outvals[(size_t)row * KSEL + kidx] = val;  // TEMP == 1
        }
        __syncthreads();
    }
}

extern "C" void kernel_launch(void* const* d_in, const int* in_sizes, int n_in,
                              void* d_out, int out_size, void* d_ws, size_t ws_size,
                              hipStream_t stream) {
    const float* gx       = (const float*)d_in[0];
    const float* cur_bias = (const float*)d_in[1];
    const float* W        = (const float*)d_in[2];
    const float* logits   = (const float*)d_in[3];

    float* out_f = (float*)d_out;                                  // filtered [B,S,K]
    int*   out_i = (int*)d_out + (size_t)Bdim * Sdim * KSEL;       // topk_ids [B,S,K]

    const int rows = Bdim * Sdim;
    topk_kernel<<<rows, TOPK_THREADS, SMEM_BYTES, stream>>>(logits, out_i);
    proposal_kernel<<<rows, 32, 0, stream>>>(gx, cur_bias, W, out_i, out_f);
}